// Encoder_79585743995180
// MI455X (gfx1250) — compile-verified
//
#include <hip/hip_runtime.h>
#include <math.h>

#define L_ 1024
#define S_ 1024
#define N_ 4
#define E_ 1024
#define H_ 16
#define M_ 512
#define D_ 64

typedef float v2f __attribute__((ext_vector_type(2)));
typedef float v8f __attribute__((ext_vector_type(8)));

__device__ __forceinline__ v8f vzero() {
  v8f z = {0.f, 0.f, 0.f, 0.f, 0.f, 0.f, 0.f, 0.f};
  return z;
}

// D = A(16x4) * B(4x16) + C, full fp32 matrix pipe
__device__ __forceinline__ v8f wmma4(v2f a, v2f b, v8f c) {
  return __builtin_amdgcn_wmma_f32_16x16x4_f32(
      /*neg_a=*/false, a, /*neg_b=*/false, b,
      /*c_mod=*/(short)0, c, /*reuse_a=*/false, /*reuse_b=*/false);
}

// sum across the 16 lanes of each half-wave (reduces over N columns of a C tile)
__device__ __forceinline__ v8f reduce_add16(v8f v) {
#pragma unroll
  for (int m = 1; m < 16; m <<= 1) {
#pragma unroll
    for (int r = 0; r < 8; ++r) v[r] += __shfl_xor(v[r], m, 32);
  }
  return v;
}

__device__ __forceinline__ v8f reduce_max16(v8f v) {
#pragma unroll
  for (int m = 1; m < 16; m <<= 1) {
#pragma unroll
    for (int r = 0; r < 8; ++r) v[r] = fmaxf(v[r], __shfl_xor(v[r], m, 32));
  }
  return v;
}

// 16x16 score tile: Q(16xD) * K(16xD)^T, both row-major stride D.
// All 32 fragment loads issued as one burst, then dual-accumulator WMMA chain.
__device__ __forceinline__ v8f qk_tile(const float* __restrict__ qrow,
                                       const float* __restrict__ krow) {
  v2f a[16], b[16];
#pragma unroll
  for (int i = 0; i < 16; ++i) {
    a[i] = *(const v2f*)(qrow + 4 * i);
    b[i] = *(const v2f*)(krow + 4 * i);
  }
  v8f ae = vzero(), ao = vzero();
#pragma unroll
  for (int i = 0; i < 16; i += 2) {
    ae = wmma4(a[i], b[i], ae);
    ao = wmma4(a[i + 1], b[i + 1], ao);
  }
  return ae + ao;
}

// 16x16 score tile vs memory keys: B[d][m] row-major with stride M (k_mem[n][h*D+d][m])
__device__ __forceinline__ v8f qkm_tile(const float* __restrict__ qrow,
                                        const float* __restrict__ KM, int m0,
                                        int lm, int lh) {
  const float* kc = KM + (size_t)(2 * lh) * M_ + m0 + lm;
  v2f a[16], b[16];
#pragma unroll
  for (int i = 0; i < 16; ++i) {
    a[i] = *(const v2f*)(qrow + 4 * i);
    b[i].x = kc[(size_t)(4 * i) * M_];
    b[i].y = kc[(size_t)(4 * i + 1) * M_];
  }
  v8f ae = vzero(), ao = vzero();
#pragma unroll
  for (int i = 0; i < 16; i += 2) {
    ae = wmma4(a[i], b[i], ae);
    ao = wmma4(a[i + 1], b[i + 1], ao);
  }
  return ae + ao;
}

// -------- C[R x Cn] = X[R x K] @ W[Cn x K]^T + bias --------
// block = 128 threads (4 waves), wave computes a 16x64 output tile.
// grid = (R/64, Cn/64). Software-pipelined: prefetch k+4 frags before k's WMMAs.
__global__ void gemm_xwt(const float* __restrict__ X, const float* __restrict__ W,
                         const float* __restrict__ bias, float* __restrict__ C,
                         int K, int Cn) {
  const int wave = threadIdx.x >> 5;
  const int lane = threadIdx.x & 31;
  const int lm = lane & 15, lh = lane >> 4;
  const int row0 = (blockIdx.x * 4 + wave) * 16;
  const int col0 = blockIdx.y * 64;
  const float* xrow = X + (size_t)(row0 + lm) * K + 2 * lh;
  const float* wrow = W + (size_t)(col0 + lm) * K + 2 * lh;
  v8f a0 = vzero(), a1 = vzero(), a2 = vzero(), a3 = vzero();
  // pipeline prologue
  v2f xa  = *(const v2f*)(xrow);
  v2f xb0 = *(const v2f*)(wrow);
  v2f xb1 = *(const v2f*)(wrow + (size_t)16 * K);
  v2f xb2 = *(const v2f*)(wrow + (size_t)32 * K);
  v2f xb3 = *(const v2f*)(wrow + (size_t)48 * K);
  for (int k = 4; k < K; k += 4) {
    v2f na  = *(const v2f*)(xrow + k);
    v2f nb0 = *(const v2f*)(wrow + k);
    v2f nb1 = *(const v2f*)(wrow + (size_t)16 * K + k);
    v2f nb2 = *(const v2f*)(wrow + (size_t)32 * K + k);
    v2f nb3 = *(const v2f*)(wrow + (size_t)48 * K + k);
    a0 = wmma4(xa, xb0, a0);
    a1 = wmma4(xa, xb1, a1);
    a2 = wmma4(xa, xb2, a2);
    a3 = wmma4(xa, xb3, a3);
    xa = na; xb0 = nb0; xb1 = nb1; xb2 = nb2; xb3 = nb3;
  }
  // pipeline epilogue
  a0 = wmma4(xa, xb0, a0);
  a1 = wmma4(xa, xb1, a1);
  a2 = wmma4(xa, xb2, a2);
  a3 = wmma4(xa, xb3, a3);
  const int orow = row0 + 8 * lh;
#pragma unroll
  for (int ct = 0; ct < 4; ++ct) {
    v8f acc = (ct == 0) ? a0 : (ct == 1) ? a1 : (ct == 2) ? a2 : a3;
    const int col = col0 + ct * 16 + lm;
    const float bv = bias[col];
#pragma unroll
    for (int r = 0; r < 8; ++r)
      C[(size_t)(orow + r) * Cn + col] = acc[r] + bv;
  }
}

// -------- RoPE + relayout (L*N, E) -> (N, H, L, D) --------
// mode 0: rope+scale (q), 1: rope (k), 2: copy (v)
__global__ void rope_reorder(const float* __restrict__ In, const float* __restrict__ Rp,
                             float* __restrict__ Oh, int mode, float scale) {
  const size_t idx = (size_t)blockIdx.x * blockDim.x + threadIdx.x; // L*N*E/2
  const int ep = (int)(idx & 511);      // E/2 = 512 pairs
  const size_t tn = idx >> 9;
  const int n = (int)(tn & 3);
  const int t = (int)(tn >> 2);
  const int e0 = ep * 2;
  const float* src = In + ((size_t)t * N_ + n) * E_ + e0;
  const float x0 = src[0], x1 = src[1];
  float y0, y1;
  if (mode == 2) {
    y0 = x0; y1 = x1;
  } else {
    const float* r = Rp + (((size_t)n * L_ + t) * E_ + e0) * 2; // [cos,sin] pairs
    const float c0 = r[0], s0 = r[1], c1 = r[2], s1 = r[3];
    y0 = (x0 * c0 - x1 * s0) * scale;
    y1 = (x1 * c1 + x0 * s1) * scale;
  }
  const int h = e0 >> 6, d = e0 & 63;
  float* dst = Oh + (((size_t)n * H_ + h) * L_ + t) * D_ + d;
  dst[0] = y0;
  dst[1] = y1;
}

// -------- column softmax stats: per (n,h,s) max_l and sum_l exp --------
// grid (N*H, S/16), block 256 (8 waves), wave w covers rows [w*128, w*128+128)
__global__ void attn_colstats(const float* __restrict__ Qh, const float* __restrict__ Kh,
                              float* __restrict__ colmax, float* __restrict__ colsum) {
  __shared__ float red[8 * 16];
  __shared__ float fin[16];
  const int nh = blockIdx.x, st = blockIdx.y;
  const float* Q  = Qh + (size_t)nh * L_ * D_;
  const float* Kp = Kh + (size_t)nh * S_ * D_;
  const int wave = threadIdx.x >> 5, lane = threadIdx.x & 31;
  const int lm = lane & 15, lh = lane >> 4;
  const float* krow = Kp + (size_t)(st * 16 + lm) * D_ + 2 * lh;
  // pass 1: column max (column = lm for both lane halves)
  float cmax = -3.0e38f;
  for (int rt = 0; rt < 8; ++rt) {
    const int row0 = wave * 128 + rt * 16;
    const float* qrow = Q + (size_t)(row0 + lm) * D_ + 2 * lh;
    v8f acc = qk_tile(qrow, krow);
#pragma unroll
    for (int r = 0; r < 8; ++r) cmax = fmaxf(cmax, acc[r]);
  }
  cmax = fmaxf(cmax, __shfl_xor(cmax, 16, 32));
  if (lane < 16) red[wave * 16 + lane] = cmax;
  __syncthreads();
  if (threadIdx.x < 16) {
    float m = red[threadIdx.x];
    for (int w = 1; w < 8; ++w) m = fmaxf(m, red[w * 16 + threadIdx.x]);
    fin[threadIdx.x] = m;
  }
  __syncthreads();
  const float colm = fin[lm];
  // pass 2: column sum of exp
  float csum = 0.f;
  for (int rt = 0; rt < 8; ++rt) {
    const int row0 = wave * 128 + rt * 16;
    const float* qrow = Q + (size_t)(row0 + lm) * D_ + 2 * lh;
    v8f acc = qk_tile(qrow, krow);
#pragma unroll
    for (int r = 0; r < 8; ++r) csum += __expf(acc[r] - colm);
  }
  csum += __shfl_xor(csum, 16, 32);
  __syncthreads();
  if (lane < 16) red[wave * 16 + lane] = csum;
  __syncthreads();
  if (threadIdx.x < 16) {
    float s = red[threadIdx.x];
    for (int w = 1; w < 8; ++w) s += red[w * 16 + threadIdx.x];
    const size_t o = (size_t)nh * S_ + st * 16 + threadIdx.x;
    colmax[o] = fin[threadIdx.x];
    colsum[o] = s;
  }
}

// -------- self-attn output: stream S tiles, w=exp(s-cmax)/csum+1e-8, row-renorm --------
// grid (N*H, L/64), block 128 (4 waves), wave owns 16 rows, output 16x64
__global__ void attn_self(const float* __restrict__ Qh, const float* __restrict__ Kh,
                          const float* __restrict__ Vh, const float* __restrict__ colmax,
                          const float* __restrict__ colsum, float* __restrict__ Out) {
  __shared__ float wstage[4][256]; // per-wave C-layout -> A-layout staging
  const int nh = blockIdx.x, lt = blockIdx.y;
  const float* Q  = Qh + (size_t)nh * L_ * D_;
  const float* Kp = Kh + (size_t)nh * S_ * D_;
  const float* V  = Vh + (size_t)nh * S_ * D_;
  const float* cm = colmax + (size_t)nh * S_;
  const float* cs = colsum + (size_t)nh * S_;
  const int wave = threadIdx.x >> 5, lane = threadIdx.x & 31;
  const int lm = lane & 15, lh = lane >> 4;
  const int row0 = lt * 64 + wave * 16;
  const float* qrow = Q + (size_t)(row0 + lm) * D_ + 2 * lh;
  float* myw = wstage[wave];
  v8f o0 = vzero(), o1 = vzero(), o2 = vzero(), o3 = vzero();
  v8f rs = vzero();
  for (int s0 = 0; s0 < S_; s0 += 16) {
    const float* krow = Kp + (size_t)(s0 + lm) * D_ + 2 * lh;
    v8f acc = qk_tile(qrow, krow);
    const float cmx = cm[s0 + lm];
    const float inv = 1.0f / cs[s0 + lm];
#pragma unroll
    for (int r = 0; r < 8; ++r) {
      const float w = __expf(acc[r] - cmx) * inv + 1e-8f;
      acc[r] = w;
      myw[(r + 8 * lh) * 16 + lm] = w;
    }
    rs += acc;
    __builtin_amdgcn_wave_barrier();
#pragma unroll
    for (int kk = 0; kk < 16; kk += 4) {
      v2f a = *(const v2f*)(myw + lm * 16 + kk + 2 * lh);
      const float* vb = V + (size_t)(s0 + kk + 2 * lh) * D_ + lm;
      v2f b0, b1, b2, b3;
      b0.x = vb[0];  b0.y = vb[D_];
      b1.x = vb[16]; b1.y = vb[16 + D_];
      b2.x = vb[32]; b2.y = vb[32 + D_];
      b3.x = vb[48]; b3.y = vb[48 + D_];
      o0 = wmma4(a, b0, o0);
      o1 = wmma4(a, b1, o1);
      o2 = wmma4(a, b2, o2);
      o3 = wmma4(a, b3, o3);
    }
    __builtin_amdgcn_wave_barrier();
  }
  rs = reduce_add16(rs); // row sums (per half: rows r / r+8)
#pragma unroll
  for (int r = 0; r < 8; ++r) {
    const float invr = 1.0f / rs[r];
    float* ob = Out + ((size_t)nh * L_ + row0 + r + 8 * lh) * D_ + lm;
    ob[0]  = o0[r] * invr;
    ob[16] = o1[r] * invr;
    ob[32] = o2[r] * invr;
    ob[48] = o3[r] * invr;
  }
}

// -------- memory attention: masked row softmax over M, then W @ Vmem --------
// grid (N*H, L/64), block 128 (4 waves), wave owns 16 rows
__global__ void attn_mem(const float* __restrict__ Qh, const float* __restrict__ Kmem,
                         const float* __restrict__ Vmem, const int* __restrict__ Mask,
                         float* __restrict__ Out) {
  __shared__ float wstage[4][256];
  const int nh = blockIdx.x, lt = blockIdx.y;
  const int n = nh >> 4, h = nh & 15;
  const float* Q  = Qh + (size_t)nh * L_ * D_;
  const float* KM = Kmem + ((size_t)n * E_ + h * D_) * M_; // [d][m], stride M
  const float* VM = Vmem + ((size_t)n * E_ + h * D_) * M_; // [d][m], stride M
  const int* mk = Mask + n * M_;
  const int wave = threadIdx.x >> 5, lane = threadIdx.x & 31;
  const int lm = lane & 15, lh = lane >> 4;
  const int row0 = lt * 64 + wave * 16;
  const float* qrow = Q + (size_t)(row0 + lm) * D_ + 2 * lh;
  // pass 1: row max (unmasked max is valid for the masked ratio)
  v8f rmax;
#pragma unroll
  for (int r = 0; r < 8; ++r) rmax[r] = -3.0e38f;
  for (int m0 = 0; m0 < M_; m0 += 16) {
    v8f acc = qkm_tile(qrow, KM, m0, lm, lh);
#pragma unroll
    for (int r = 0; r < 8; ++r) rmax[r] = fmaxf(rmax[r], acc[r]);
  }
  rmax = reduce_max16(rmax);
  // pass 2: masked row sum of exp
  v8f rsum = vzero();
  for (int m0 = 0; m0 < M_; m0 += 16) {
    v8f acc = qkm_tile(qrow, KM, m0, lm, lh);
    const float mf = (float)mk[m0 + lm];
#pragma unroll
    for (int r = 0; r < 8; ++r) rsum[r] += mf * __expf(acc[r] - rmax[r]);
  }
  rsum = reduce_add16(rsum);
  v8f rinv;
#pragma unroll
  for (int r = 0; r < 8; ++r) rinv[r] = 1.0f / rsum[r];
  // pass 3: normalized masked weights, PV via WMMA
  v8f o0 = vzero(), o1 = vzero(), o2 = vzero(), o3 = vzero();
  float* myw = wstage[wave];
  for (int m0 = 0; m0 < M_; m0 += 16) {
    v8f acc = qkm_tile(qrow, KM, m0, lm, lh);
    const float mf = (float)mk[m0 + lm];
#pragma unroll
    for (int r = 0; r < 8; ++r)
      myw[(r + 8 * lh) * 16 + lm] = mf * __expf(acc[r] - rmax[r]) * rinv[r];
    __builtin_amdgcn_wave_barrier();
#pragma unroll
    for (int kk = 0; kk < 16; kk += 4) {
      v2f a = *(const v2f*)(myw + lm * 16 + kk + 2 * lh);
      const float* vb = VM + (size_t)lm * M_ + m0 + kk + 2 * lh;
      v2f b0 = *(const v2f*)(vb);
      v2f b1 = *(const v2f*)(vb + (size_t)16 * M_);
      v2f b2 = *(const v2f*)(vb + (size_t)32 * M_);
      v2f b3 = *(const v2f*)(vb + (size_t)48 * M_);
      o0 = wmma4(a, b0, o0);
      o1 = wmma4(a, b1, o1);
      o2 = wmma4(a, b2, o2);
      o3 = wmma4(a, b3, o3);
    }
    __builtin_amdgcn_wave_barrier();
  }
#pragma unroll
  for (int r = 0; r < 8; ++r) {
    float* ob = Out + ((size_t)nh * L_ + row0 + r + 8 * lh) * D_ + lm;
    ob[0]  = o0[r];
    ob[16] = o1[r];
    ob[32] = o2[r];
    ob[48] = o3[r];
  }
}

// -------- sigmoid gate blend + relayout (N,H,L,D) -> (L*N, E) --------
__global__ void combine_gate(const float* __restrict__ Os, const float* __restrict__ Om,
                             const float* __restrict__ gate, float* __restrict__ Comb) {
  const size_t i = (size_t)blockIdx.x * blockDim.x + threadIdx.x; // N*H*L*D
  const int d = (int)(i & 63);
  const size_t t = i >> 6; // nh*L + l
  const int l = (int)(t & (L_ - 1));
  const int nhh = (int)(t >> 10);
  const int h = nhh & 15, n = nhh >> 4;
  const float g = 1.0f / (1.0f + __expf(-gate[h]));
  const float v = g * Om[i] + (1.0f - g) * Os[i];
  Comb[((size_t)l * N_ + n) * E_ + h * D_ + d] = v;
}

extern "C" void kernel_launch(void* const* d_in, const int* in_sizes, int n_in,
                              void* d_out, int out_size, void* d_ws, size_t ws_size,
                              hipStream_t stream) {
  (void)in_sizes; (void)n_in; (void)out_size; (void)ws_size;
  const float* query = (const float*)d_in[0];
  const float* key   = (const float*)d_in[1];
  const float* value = (const float*)d_in[2];
  const float* ipw   = (const float*)d_in[3];
  const float* ipb   = (const float*)d_in[4];
  const float* opw   = (const float*)d_in[5];
  const float* opb   = (const float*)d_in[6];
  const float* qp    = (const float*)d_in[7];
  const float* kvp   = (const float*)d_in[8];
  const float* kmem  = (const float*)d_in[9];
  const float* vmem  = (const float*)d_in[10];
  const float* gate  = (const float*)d_in[11];
  const int*   mask  = (const int*)d_in[12];
  float* out = (float*)d_out;

  float* ws = (float*)d_ws;
  const size_t SZ = (size_t)L_ * N_ * E_; // 4M floats
  float* bufA = ws;           // gemm scratch -> self-attn out
  float* bufB = ws + SZ;      // mem-attn out
  float* qh   = ws + 2 * SZ;  // (N,H,L,D); later reused for gated combine
  float* kh   = ws + 3 * SZ;
  float* vh   = ws + 4 * SZ;
  float* cmx  = ws + 5 * SZ;                        // N*H*S
  float* csm  = cmx + (size_t)N_ * H_ * S_;         // N*H*S

  const dim3 ggrid(64, 16); // R/64=64, Cn/64=16
  const dim3 gblk(128);

  // Q/K/V projections + RoPE + head relayout (bufA reused as GEMM temp)
  gemm_xwt<<<ggrid, gblk, 0, stream>>>(query, ipw, ipb, bufA, E_, E_);
  rope_reorder<<<8192, 256, 0, stream>>>(bufA, qp, qh, 0, 0.125f); // D^-0.5
  gemm_xwt<<<ggrid, gblk, 0, stream>>>(key, ipw + (size_t)E_ * E_, ipb + E_, bufA, E_, E_);
  rope_reorder<<<8192, 256, 0, stream>>>(bufA, kvp, kh, 1, 1.0f);
  gemm_xwt<<<ggrid, gblk, 0, stream>>>(value, ipw + 2 * (size_t)E_ * E_, ipb + 2 * E_, bufA, E_, E_);
  rope_reorder<<<8192, 256, 0, stream>>>(bufA, qp /*unused*/, vh, 2, 1.0f);

  // self-attention (column softmax stats, then streamed output)
  attn_colstats<<<dim3(64, 64), 256, 0, stream>>>(qh, kh, cmx, csm);
  attn_self<<<dim3(64, 16), gblk, 0, stream>>>(qh, kh, vh, cmx, csm, bufA);

  // memory attention
  attn_mem<<<dim3(64, 16), gblk, 0, stream>>>(qh, kmem, vmem, mask, bufB);

  // gate blend into (L*N, E) rows, then output projection
  combine_gate<<<16384, 256, 0, stream>>>(bufA, bufB, gate, qh);
  gemm_xwt<<<ggrid, gblk, 0, stream>>>(qh, opw, opb, out, E_, E_);
}